// Contour_to_mask_34763465294599
// MI455X (gfx1250) — compile-verified
//
#include <hip/hip_runtime.h>
#include <math.h>

typedef float v2f __attribute__((ext_vector_type(2)));
typedef float v8f __attribute__((ext_vector_type(8)));

#define KSIGN  100000.0f
#define EPSC   1e-5f
#define INV2PI 0.15915494309189535f
#define PI_F   3.14159265358979323846f

// ---- branchless device transcendentals ------------------------------------

static __device__ __forceinline__ float dev_tanh(float x) {
#if __has_builtin(__builtin_amdgcn_tanhf)
    return __builtin_amdgcn_tanhf(x);      // v_tanh_f32 (CDNA5 TRANS op)
#else
    // branchless fallback: tanh(|x|) = 1 - 2/(exp(2|x|)+1); exp->inf saturates to 1
    float ax = fabsf(x);
    float e  = __expf(2.0f * ax);          // v_exp_f32 based, branchless
    float t  = 1.0f - 2.0f / (e + 1.0f);
    return copysignf(t, x);
#endif
}

static __device__ __forceinline__ float dev_rsqrt(float x) {
#if __has_builtin(__builtin_amdgcn_rsqf)
    return __builtin_amdgcn_rsqf(x);       // v_rsq_f32
#else
    return rsqrtf(x);
#endif
}

static __device__ __forceinline__ float dev_sqrt(float x) {
#if __has_builtin(__builtin_amdgcn_sqrtf)
    return __builtin_amdgcn_sqrtf(x);      // v_sqrt_f32
#else
    return sqrtf(x);
#endif
}

// Branchless acos, |err| <= ~7e-5 rad (Abramowitz-Stegun 4.4.45):
// acos(a) = sqrt(1-a)*P(a) for a in [0,1];  acos(x<0) = pi - acos(-x)
static __device__ __forceinline__ float dev_acos(float x) {
    float ax = fabsf(x);
    float p  = fmaf(ax, -0.0187293f, 0.0742610f);
    p        = fmaf(ax, p, -0.2121144f);
    p        = fmaf(ax, p, 1.5707288f);
    float r  = dev_sqrt(fmaxf(1.0f - ax, 0.0f)) * p;
    return (x >= 0.0f) ? r : (PI_F - r);   // v_cndmask, no exec branch
}

// ---------------------------------------------------------------------------
// One wave handles 16-pixel tiles of one contour.
// WMMA 16x16x4 f32: A = 16 edges x 4 coeffs, B = 4 pixel-features x 16 pixels,
// D[edge][pixel]. Four quantities (cross, dot, |d|^2, |rd|^2) x 4 edge groups
// = 16 WMMAs per tile; branchless tanh/acos on VALU (co-executes with WMMA);
// edge reduction = 32 in-lane FMAs + one shfl_xor(16); coalesced 16-lane store.
__global__ __launch_bounds__(256) void contour_mask_wmma(
    const float* __restrict__ contour, float* __restrict__ out,
    int S, int tprShift)
{
    const int bn   = blockIdx.x;            // contour index (B*N)
    const int lane = threadIdx.x & 31;
    const int wave = threadIdx.x >> 5;
    const int nwav = blockDim.x >> 5;
    const int col  = lane & 15;
    const bool hi  = lane >= 16;
    const float inv = 1.0f / (float)S;

    const float* __restrict__ c = contour + (size_t)bn * 64u * 2u;

    // ---- Build per-contour A matrices (resident in VGPRs) -----------------
    // Feature order: (1, p0, p1, r2). A lane layout (16x4 f32):
    // lanes 0-15 hold {coef0, coef1} of edge row=lane; lanes 16-31 hold {coef2, coef3}.
    v2f A[4][4];
#pragma unroll
    for (int g = 0; g < 4; ++g) {
        const int v  = g * 16 + col;
        const int vn = (v + 1) & 63;
        const float c0 = c[2 * v],  c1 = c[2 * v + 1];
        const float n0 = c[2 * vn], n1 = c[2 * vn + 1];
        // cross = Kc + Lc*p0 + Mc*p1 + 0*r2
        const float Kc = c1 * n0 - c0 * n1;
        const float Lc = n1 - c1;
        const float Mc = c0 - n0;
        // dot  = S0 + Sx*p0 + Sy*p1 + 1*r2
        const float S0 = c0 * n0 + c1 * n1;
        const float Sx = -(c0 + n0);
        const float Sy = -(c1 + n1);
        // |d|^2  = D0 + Dx*p0 + Dy*p1 + 1*r2 ; |rd|^2 with next vertex
        const float D0 = c0 * c0 + c1 * c1, Dx = -2.0f * c0, Dy = -2.0f * c1;
        const float R0 = n0 * n0 + n1 * n1, Rx = -2.0f * n0, Ry = -2.0f * n1;

        v2f a0lo = {Kc, Lc}, a0hi = {Mc, 0.0f};
        v2f a1lo = {S0, Sx}, a1hi = {Sy, 1.0f};
        v2f a2lo = {D0, Dx}, a2hi = {Dy, 1.0f};
        v2f a3lo = {R0, Rx}, a3hi = {Ry, 1.0f};
        A[g][0] = hi ? a0hi : a0lo;
        A[g][1] = hi ? a1hi : a1lo;
        A[g][2] = hi ? a2hi : a2lo;
        A[g][3] = hi ? a3hi : a3lo;
    }

    // ---- Sweep pixel tiles (tilesPerRow is a power of two) ----------------
    const int tprMask    = (1 << tprShift) - 1;
    const int tilesTotal = S << tprShift;
    for (int t = blockIdx.y * nwav + wave; t < tilesTotal; t += gridDim.y * nwav) {
        const int i  = t >> tprShift;
        const int j0 = (t & tprMask) << 4;
        const float p0 = (float)i * inv;
        const float p1 = (float)(j0 + col) * inv;
        const float r2 = p0 * p0 + p1 * p1;
        // B layout (4x16 f32): lanes 0-15 {f0,f1}, lanes 16-31 {f2,f3}, col = lane%16
        v2f blo = {1.0f, p0}, bhi = {p1, r2};
        const v2f b = hi ? bhi : blo;

        float acc = 0.0f;
#pragma unroll
        for (int g = 0; g < 4; ++g) {
            const v8f z = {};
            v8f dc = __builtin_amdgcn_wmma_f32_16x16x4_f32(false, A[g][0], false, b, (short)0, z, false, false);
            v8f ds = __builtin_amdgcn_wmma_f32_16x16x4_f32(false, A[g][1], false, b, (short)0, z, false, false);
            v8f dn = __builtin_amdgcn_wmma_f32_16x16x4_f32(false, A[g][2], false, b, (short)0, z, false, false);
            v8f dr = __builtin_amdgcn_wmma_f32_16x16x4_f32(false, A[g][3], false, b, (short)0, z, false, false);
#pragma unroll
            for (int r = 0; r < 8; ++r) {
                const float cross = dc[r];
                const float dotv  = ds[r];
                const float nd2   = dn[r];
                const float nrd2  = dr[r];
                const float sgn   = dev_tanh(KSIGN * cross);
                const float rinv  = dev_rsqrt(nd2 * nrd2);
                const float ca    = fminf(fmaxf(dotv * rinv, -1.0f + EPSC), 1.0f - EPSC);
                acc = fmaf(sgn, dev_acos(ca), acc);
            }
        }
        // Edge rows 0-7 live in lanes 0-15, rows 8-15 in lanes 16-31 (same pixel col):
        acc += __shfl_xor(acc, 16, 32);
        const float valp = fminf(fabsf(acc) * INV2PI, 1.0f);
        if (!hi) {
            out[(size_t)bn * S * S + (size_t)i * S + (size_t)(j0 + col)] = valp;
        }
    }
}

extern "C" void kernel_launch(void* const* d_in, const int* in_sizes, int n_in,
                              void* d_out, int out_size, void* d_ws, size_t ws_size,
                              hipStream_t stream) {
    (void)n_in; (void)d_ws; (void)ws_size;
    const float* contour = (const float*)d_in[0];
    float* out = (float*)d_out;

    int bn = in_sizes[0] / (64 * 2);     // B*N contours, KV=64, 2 coords
    if (bn <= 0) bn = 16;
    int S = (int)(sqrt((double)out_size / (double)bn) + 0.5);
    if (S < 16) S = 16;

    int tilesPerRow = S >> 4;
    if (tilesPerRow < 1) tilesPerRow = 1;
    int tprShift = 0;
    while ((1 << tprShift) < tilesPerRow) ++tprShift;

    dim3 grid((unsigned)bn, 64u);
    contour_mask_wmma<<<grid, 256, 0, stream>>>(contour, out, S, tprShift);
}